// Rigid_Loss_56796647522829
// MI455X (gfx1250) — compile-verified
//
#include <hip/hip_runtime.h>
#include <math.h>

#define REPS 1e-6f

typedef __attribute__((ext_vector_type(2))) float v2f;
typedef __attribute__((ext_vector_type(8))) float v8f;

// Exact full-wave32 f32 sum using V_WMMA_F32_16X16X4_F32.
// A layout (ISA 7.12.2, 32-bit A 16x4): lanes 0-15 -> rows m, VGPR0=K0, VGPR1=K1;
// lanes 16-31 -> rows m, VGPR0=K2, VGPR1=K3.  With A[m][0]=x[m], A[m][2]=x[m+16],
// A[*][1]=A[*][3]=0 and B=ones(4x16):  D[m][n] = x[m] + x[m+16]  (all n equal).
// D layout: lane n holds rows 0-7 (lanes<16) or 8-15 (lanes>=16) in its 8 VGPRs,
// so a per-lane sum of the 8 VGPRs + one xor-16 shuffle yields the full sum in
// every lane.  Must be executed with full EXEC.
__device__ __forceinline__ float wave_sum_wmma(float x) {
    v2f a; a[0] = x;    a[1] = 0.0f;
    v2f b; b[0] = 1.0f; b[1] = 1.0f;
    v8f c = {};
    v8f d = __builtin_amdgcn_wmma_f32_16x16x4_f32(
        /*neg_a=*/false, a, /*neg_b=*/false, b,
        /*c_mod=*/(short)0, c, /*reuse_a=*/false, /*reuse_b=*/false);
    float s = ((d[0] + d[1]) + (d[2] + d[3])) + ((d[4] + d[5]) + (d[6] + d[7]));
    s += __shfl_xor(s, 16, 32);
    return s;
}

__global__ __launch_bounds__(256) void rigid_loss_main(
    const float* __restrict__ trg,   // new_verts_coords, V*3
    const float* __restrict__ src,   // verts_src,        V*3
    const int*   __restrict__ nbi,   // neighborhood_indices, V*16 (int32)
    const float* __restrict__ nbw,   // neighborhood_weights, V*16
    float* __restrict__ partial,     // 2 floats per block
    int V)
{
    const int tid = threadIdx.x;
    const int v   = blockIdx.x * 256 + tid;
    const bool active = (v < V);

    float num = 0.0f, den = 0.0f;

    if (active) {
        // ---- centers ----
        const float csx = src[3 * v + 0], csy = src[3 * v + 1], csz = src[3 * v + 2];
        const float ctx = trg[3 * v + 0], cty = trg[3 * v + 1], ctz = trg[3 * v + 2];

        // ---- coalesced b128 loads of indices + weights ----
        int   idx[16];
        float w[16];
        const int4*   ni4 = (const int4*)(nbi + (size_t)v * 16);
        const float4* nw4 = (const float4*)(nbw + (size_t)v * 16);
        #pragma unroll
        for (int q = 0; q < 4; ++q) {
            int4 t4 = ni4[q];
            idx[4 * q + 0] = t4.x; idx[4 * q + 1] = t4.y;
            idx[4 * q + 2] = t4.z; idx[4 * q + 3] = t4.w;
        }
        #pragma unroll
        for (int q = 0; q < 4; ++q) {
            float4 f4 = nw4[q];
            w[4 * q + 0] = f4.x; w[4 * q + 1] = f4.y;
            w[4 * q + 2] = f4.z; w[4 * q + 3] = f4.w;
            den += (f4.x + f4.y) + (f4.z + f4.w);
        }

        // ---- pass 1: 3x3 cross-covariance  M[i][j] = sum_d s_i t_j ----
        float m00 = 0, m01 = 0, m02 = 0, m10 = 0, m11 = 0, m12 = 0, m20 = 0, m21 = 0, m22 = 0;
        #pragma unroll
        for (int d_ = 0; d_ < 16; ++d_) {
            const int j = idx[d_];
            const float sx = src[3 * j + 0] - csx;
            const float sy = src[3 * j + 1] - csy;
            const float sz = src[3 * j + 2] - csz;
            const float tx = trg[3 * j + 0] - ctx;
            const float ty = trg[3 * j + 1] - cty;
            const float tz = trg[3 * j + 2] - ctz;
            m00 = fmaf(sx, tx, m00); m01 = fmaf(sx, ty, m01); m02 = fmaf(sx, tz, m02);
            m10 = fmaf(sy, tx, m10); m11 = fmaf(sy, ty, m11); m12 = fmaf(sy, tz, m12);
            m20 = fmaf(sz, tx, m20); m21 = fmaf(sz, ty, m21); m22 = fmaf(sz, tz, m22);
        }
        m00 += REPS; m11 += REPS; m22 += REPS;

        // ---- scaled Newton polar iteration:  Q <- 0.5*(zeta*Q + (1/zeta)*Q^-T)
        //      Q^-T = cof(Q)/det;  zeta = sqrt(||Q^-1||_F / ||Q||_F)         ----
        float q00 = m00, q01 = m01, q02 = m02;
        float q10 = m10, q11 = m11, q12 = m12;
        float q20 = m20, q21 = m21, q22 = m22;
        #pragma unroll
        for (int it = 0; it < 8; ++it) {
            const float c00 = fmaf(q11, q22, -q12 * q21);
            const float c01 = fmaf(q12, q20, -q10 * q22);
            const float c02 = fmaf(q10, q21, -q11 * q20);
            const float c10 = fmaf(q02, q21, -q01 * q22);
            const float c11 = fmaf(q00, q22, -q02 * q20);
            const float c12 = fmaf(q01, q20, -q00 * q21);
            const float c20 = fmaf(q01, q12, -q02 * q11);
            const float c21 = fmaf(q02, q10, -q00 * q12);
            const float c22 = fmaf(q00, q11, -q01 * q10);
            float det = q00 * c00 + q01 * c01 + q02 * c02;
            const float adet = fmaxf(fabsf(det), 1e-20f);
            det = (det < 0.0f) ? -adet : adet;
            float sq = q00*q00 + q01*q01 + q02*q02 + q10*q10 + q11*q11
                     + q12*q12 + q20*q20 + q21*q21 + q22*q22;
            float sc = c00*c00 + c01*c01 + c02*c02 + c10*c10 + c11*c11
                     + c12*c12 + c20*c20 + c21*c21 + c22*c22;
            sq = fmaxf(sq, 1e-30f);
            sc = fmaxf(sc, 1e-30f);
            const float zeta = sqrtf(sqrtf(sc / sq) / adet);
            const float hz = 0.5f * zeta;
            const float hi = 0.5f / (zeta * det);
            q00 = fmaf(hz, q00, hi * c00); q01 = fmaf(hz, q01, hi * c01); q02 = fmaf(hz, q02, hi * c02);
            q10 = fmaf(hz, q10, hi * c10); q11 = fmaf(hz, q11, hi * c11); q12 = fmaf(hz, q12, hi * c12);
            q20 = fmaf(hz, q20, hi * c20); q21 = fmaf(hz, q21, hi * c21); q22 = fmaf(hz, q22, hi * c22);
        }

        // ---- pass 2: re-gather (L2 hits), Y_hat = Q*s, accumulate w*||Q*s - t|| ----
        #pragma unroll
        for (int d_ = 0; d_ < 16; ++d_) {
            const int j = idx[d_];
            const float sx = src[3 * j + 0] - csx;
            const float sy = src[3 * j + 1] - csy;
            const float sz = src[3 * j + 2] - csz;
            const float tx = trg[3 * j + 0] - ctx;
            const float ty = trg[3 * j + 1] - cty;
            const float tz = trg[3 * j + 2] - ctz;
            const float yx = fmaf(q00, sx, fmaf(q01, sy, q02 * sz));
            const float yy = fmaf(q10, sx, fmaf(q11, sy, q12 * sz));
            const float yz = fmaf(q20, sx, fmaf(q21, sy, q22 * sz));
            const float ex = yx - tx, ey = yy - ty, ez = yz - tz;
            num = fmaf(sqrtf(fmaf(ex, ex, fmaf(ey, ey, ez * ez))), w[d_], num);
        }
    }

    // ---- full-EXEC wave reduction via WMMA, then 8-wave LDS combine ----
    const float wnum = wave_sum_wmma(num);
    const float wden = wave_sum_wmma(den);

    __shared__ float snum[8], sden[8];
    const int wave = tid >> 5;
    if ((tid & 31) == 0) { snum[wave] = wnum; sden[wave] = wden; }
    __syncthreads();
    if (tid == 0) {
        float a = 0.0f, b = 0.0f;
        #pragma unroll
        for (int i = 0; i < 8; ++i) { a += snum[i]; b += sden[i]; }
        partial[2 * (size_t)blockIdx.x + 0] = a;
        partial[2 * (size_t)blockIdx.x + 1] = b;
    }
}

__global__ __launch_bounds__(256) void rigid_loss_finalize(
    const float* __restrict__ partial, int nblk, float* __restrict__ out)
{
    const int tid = threadIdx.x;
    float num = 0.0f, den = 0.0f;
    for (int i = tid; i < nblk; i += 256) {
        num += partial[2 * (size_t)i + 0];
        den += partial[2 * (size_t)i + 1];
    }
    const float wnum = wave_sum_wmma(num);
    const float wden = wave_sum_wmma(den);

    __shared__ float snum[8], sden[8];
    const int wave = tid >> 5;
    if ((tid & 31) == 0) { snum[wave] = wnum; sden[wave] = wden; }
    __syncthreads();
    if (tid == 0) {
        float a = 0.0f, b = 0.0f;
        #pragma unroll
        for (int i = 0; i < 8; ++i) { a += snum[i]; b += sden[i]; }
        out[0] = a / (b + REPS);
    }
}

extern "C" void kernel_launch(void* const* d_in, const int* in_sizes, int n_in,
                              void* d_out, int out_size, void* d_ws, size_t ws_size,
                              hipStream_t stream) {
    const float* trg = (const float*)d_in[0];   // new_verts_coords
    const float* src = (const float*)d_in[1];   // verts_src
    const int*   nbi = (const int*)d_in[2];     // neighborhood_indices (int32)
    const float* nbw = (const float*)d_in[3];   // neighborhood_weights
    float* out = (float*)d_out;
    float* partial = (float*)d_ws;

    const int V = in_sizes[0] / 3;
    const int nblk = (V + 255) / 256;

    rigid_loss_main<<<nblk, 256, 0, stream>>>(trg, src, nbi, nbw, partial, V);
    rigid_loss_finalize<<<1, 256, 0, stream>>>(partial, nblk, out);
}